// PolylineEncoder_25503515803955
// MI455X (gfx1250) — compile-verified
//
#include <hip/hip_runtime.h>
#include <hip/hip_bf16.h>

namespace {

constexpr int  kT   = 16, kB = 32, kN = 4096;
constexpr int  kNT  = 20, kF = 25, kH = 8, kHID = 128;
constexpr int  kTB  = kT * kB;                    // 512 (t,b) slots
constexpr long kTBN = (long)kTB * kN;             // 2,097,152 points
constexpr int  kPoolW = kNT * kH;                 // 160 pooled feats per slot
constexpr int  kBlk = 256;
constexpr int  kBlksPerTB = kN / kBlk;            // 16 blocks per (t,b) slot

typedef __attribute__((ext_vector_type(2))) float v2f;
typedef __attribute__((ext_vector_type(8))) float v8f;

__device__ __forceinline__ float relu(float v) { return v > 0.f ? v : 0.f; }

// ---------------------------------------------------------------------------
// Workspace pool init (harness does not re-zero ws between replays)
// ---------------------------------------------------------------------------
__global__ void k_zero(unsigned* __restrict__ p, int n) {
  int i = blockIdx.x * blockDim.x + threadIdx.x;
  if (i < n) p[i] = 0u;
}

// ---------------------------------------------------------------------------
// Front half of the point MLP: x(25) -> h3(8), argmax type, valid flag.
// Weights come from LDS (uniform-address broadcast reads).
// ---------------------------------------------------------------------------
__device__ __forceinline__ void point_front(
    const float* __restrict__ x,
    const float* __restrict__ sW1, const float* __restrict__ sB1,
    const float* __restrict__ sW2, const float* __restrict__ sB2,
    const float* __restrict__ sW3, const float* __restrict__ sB3,
    float h3[kH], int& ptype, bool& valid) {
  float xf[kF];
#pragma unroll
  for (int i = 0; i < kF; i++) xf[i] = x[i];

  // argmax over one-hot slice [4, 24): strict '>' keeps first occurrence
  int t = 0; float best = xf[4];
#pragma unroll
  for (int k = 1; k < kNT; k++) {
    float v = xf[4 + k];
    if (v > best) { best = v; t = k; }
  }
  ptype = t;
  valid = (xf[kF - 1] != 0.0f);

  float h1[kH];
#pragma unroll
  for (int o = 0; o < kH; o++) h1[o] = sB1[o];
#pragma unroll
  for (int k = 0; k < kF; k++) {
    float xk = xf[k];
#pragma unroll
    for (int o = 0; o < kH; o++) h1[o] = fmaf(xk, sW1[k * kH + o], h1[o]);
  }
#pragma unroll
  for (int o = 0; o < kH; o++) h1[o] = relu(h1[o]);

  float h2[kH];
#pragma unroll
  for (int o = 0; o < kH; o++) h2[o] = sB2[o];
#pragma unroll
  for (int k = 0; k < kH; k++) {
    float xk = h1[k];
#pragma unroll
    for (int o = 0; o < kH; o++) h2[o] = fmaf(xk, sW2[k * kH + o], h2[o]);
  }
#pragma unroll
  for (int o = 0; o < kH; o++) h2[o] = relu(h2[o]);

#pragma unroll
  for (int o = 0; o < kH; o++) h3[o] = sB3[o];
#pragma unroll
  for (int k = 0; k < kH; k++) {
    float xk = h2[k];
#pragma unroll
    for (int o = 0; o < kH; o++) h3[o] = fmaf(xk, sW3[k * kH + o], h3[o]);
  }
#pragma unroll
  for (int o = 0; o < kH; o++) h3[o] = valid ? relu(h3[o]) : 0.0f;
}

// ---------------------------------------------------------------------------
// Phase 1: per-point front MLP + first segment-max pool.
// One block = 256 consecutive points inside a single (t,b) slot.
// Pool values are ReLU'd (>=0) so float max == uint max on the raw bits.
// ---------------------------------------------------------------------------
template <bool STORE_H>
__global__ __launch_bounds__(kBlk) void k_phase1(
    const float* __restrict__ pts,
    const float* __restrict__ W1, const float* __restrict__ b1,
    const float* __restrict__ W2, const float* __restrict__ b2,
    const float* __restrict__ W3, const float* __restrict__ b3,
    float* __restrict__ h3buf, unsigned char* __restrict__ typebuf,
    unsigned* __restrict__ gpool1) {
  __shared__ __align__(16) float sW1[kF * kH];
  __shared__ __align__(16) float sW2[kH * kH];
  __shared__ __align__(16) float sW3[kH * kH];
  __shared__ float sB1[kH], sB2[kH], sB3[kH];
  __shared__ unsigned spool[kPoolW];

  int tid = threadIdx.x;
  for (int i = tid; i < kF * kH; i += kBlk) sW1[i] = W1[i];
  for (int i = tid; i < kH * kH; i += kBlk) { sW2[i] = W2[i]; sW3[i] = W3[i]; }
  if (tid < kH) { sB1[tid] = b1[tid]; sB2[tid] = b2[tid]; sB3[tid] = b3[tid]; }
  if (tid < kPoolW) spool[tid] = 0u;
  __syncthreads();

  long idx = (long)blockIdx.x * kBlk + tid;
  float h3[kH]; int ptype; bool valid;
  point_front(pts + idx * kF, sW1, sB1, sW2, sB2, sW3, sB3, h3, ptype, valid);

  if (STORE_H) {
    // idx*8 floats -> 32B aligned: two b128 stores
    float4 lo = make_float4(h3[0], h3[1], h3[2], h3[3]);
    float4 hi = make_float4(h3[4], h3[5], h3[6], h3[7]);
    float4* dst = (float4*)(h3buf + idx * kH);
    dst[0] = lo; dst[1] = hi;
    typebuf[idx] = valid ? (unsigned char)ptype : (unsigned char)0xFF;
  }

  if (valid) {
#pragma unroll
    for (int o = 0; o < kH; o++)
      atomicMax(&spool[ptype * kH + o], __float_as_uint(h3[o]));
  }
  __syncthreads();

  int tb = blockIdx.x / kBlksPerTB;
  if (tid < kPoolW) atomicMax(&gpool1[tb * kPoolW + tid], spool[tid]);
}

// ---------------------------------------------------------------------------
// Phase 2: concat(h3, pooled[type]) -> layers 4..6 -> second segment-max pool.
// Invalid points contribute exactly the 0 floor, so they are skipped.
// RECOMPUTE path re-derives h3 from points when ws can't hold the h3 cache.
// ---------------------------------------------------------------------------
template <bool RECOMPUTE>
__global__ __launch_bounds__(kBlk) void k_phase2(
    const float* __restrict__ pts,
    const float* __restrict__ W1, const float* __restrict__ b1,
    const float* __restrict__ W2, const float* __restrict__ b2,
    const float* __restrict__ W3, const float* __restrict__ b3,
    const float* __restrict__ W4, const float* __restrict__ b4,
    const float* __restrict__ W5, const float* __restrict__ b5,
    const float* __restrict__ W6, const float* __restrict__ b6,
    const float* __restrict__ h3buf, const unsigned char* __restrict__ typebuf,
    const unsigned* __restrict__ gpool1, unsigned* __restrict__ gpool2) {
  __shared__ __align__(16) float sW1[kF * kH];
  __shared__ __align__(16) float sW2[kH * kH];
  __shared__ __align__(16) float sW3[kH * kH];
  __shared__ __align__(16) float sW4[2 * kH * kH];
  __shared__ __align__(16) float sW5[kH * kH];
  __shared__ __align__(16) float sW6[kH * kH];
  __shared__ float sB1[kH], sB2[kH], sB3[kH], sB4[kH], sB5[kH], sB6[kH];
  __shared__ unsigned spool[kPoolW];

  int tid = threadIdx.x;
  if (RECOMPUTE) {
    for (int i = tid; i < kF * kH; i += kBlk) sW1[i] = W1[i];
    for (int i = tid; i < kH * kH; i += kBlk) { sW2[i] = W2[i]; sW3[i] = W3[i]; }
    if (tid < kH) { sB1[tid] = b1[tid]; sB2[tid] = b2[tid]; sB3[tid] = b3[tid]; }
  }
  for (int i = tid; i < 2 * kH * kH; i += kBlk) sW4[i] = W4[i];
  for (int i = tid; i < kH * kH; i += kBlk) { sW5[i] = W5[i]; sW6[i] = W6[i]; }
  if (tid < kH) { sB4[tid] = b4[tid]; sB5[tid] = b5[tid]; sB6[tid] = b6[tid]; }
  if (tid < kPoolW) spool[tid] = 0u;
  __syncthreads();

  long idx = (long)blockIdx.x * kBlk + tid;
  int tb = blockIdx.x / kBlksPerTB;

  float h3[kH]; int ptype; bool valid;
  if (RECOMPUTE) {
    point_front(pts + idx * kF, sW1, sB1, sW2, sB2, sW3, sB3, h3, ptype, valid);
  } else {
    unsigned char tc = typebuf[idx];
    valid = (tc != 0xFF);
    ptype = valid ? (int)tc : 0;
    const float4* src = (const float4*)(h3buf + idx * kH);
    float4 lo = src[0], hi = src[1];
    h3[0] = lo.x; h3[1] = lo.y; h3[2] = lo.z; h3[3] = lo.w;
    h3[4] = hi.x; h3[5] = hi.y; h3[6] = hi.z; h3[7] = hi.w;
  }

  if (valid) {
    float pl[kH];
    const unsigned* pp = gpool1 + tb * kPoolW + ptype * kH;
#pragma unroll
    for (int o = 0; o < kH; o++) pl[o] = __uint_as_float(pp[o]);

    float h4[kH];
#pragma unroll
    for (int o = 0; o < kH; o++) h4[o] = sB4[o];
#pragma unroll
    for (int k = 0; k < kH; k++) {
      float a = h3[k], p = pl[k];
#pragma unroll
      for (int o = 0; o < kH; o++) {
        h4[o] = fmaf(a, sW4[k * kH + o], h4[o]);
        h4[o] = fmaf(p, sW4[(kH + k) * kH + o], h4[o]);
      }
    }
#pragma unroll
    for (int o = 0; o < kH; o++) h4[o] = relu(h4[o]);

    float h5[kH];
#pragma unroll
    for (int o = 0; o < kH; o++) h5[o] = sB5[o];
#pragma unroll
    for (int k = 0; k < kH; k++) {
      float a = h4[k];
#pragma unroll
      for (int o = 0; o < kH; o++) h5[o] = fmaf(a, sW5[k * kH + o], h5[o]);
    }
#pragma unroll
    for (int o = 0; o < kH; o++) h5[o] = relu(h5[o]);

    float h6[kH];
#pragma unroll
    for (int o = 0; o < kH; o++) h6[o] = sB6[o];
#pragma unroll
    for (int k = 0; k < kH; k++) {
      float a = h5[k];
#pragma unroll
      for (int o = 0; o < kH; o++) h6[o] = fmaf(a, sW6[k * kH + o], h6[o]);
    }
#pragma unroll
    for (int o = 0; o < kH; o++) {
      h6[o] = relu(h6[o]);
      atomicMax(&spool[ptype * kH + o], __float_as_uint(h6[o]));
    }
  }
  __syncthreads();

  if (tid < kPoolW) atomicMax(&gpool2[tb * kPoolW + tid], spool[tid]);
}

// ---------------------------------------------------------------------------
// Phase 3: out(512x128) = relu(P2(512x160) @ Wout(160x128) + bout)
// One wave per 16x16 output tile, V_WMMA_F32_16X16X4_F32, K unrolled 40x.
// A-frag (16x4 f32): lanes 0-15 hold M=lane, VGPR0/1 = K=k0,k0+1;
//                    lanes 16-31 hold K=k0+2,k0+3. B-frag mirrored.
// ---------------------------------------------------------------------------
__global__ __launch_bounds__(32) void k_out_gemm(
    const unsigned* __restrict__ gpool2, const float* __restrict__ Wout,
    const float* __restrict__ bout, float* __restrict__ out) {
  const float* P2 = (const float*)gpool2;  // 512 x 160, all values >= 0
  int lane = threadIdx.x;
  int half = lane >> 4;
  int l16 = lane & 15;
  int mBase = blockIdx.y * 16;  // over 512 rows
  int nBase = blockIdx.x * 16;  // over 128 cols

  v8f acc = {};
#pragma unroll
  for (int k0 = 0; k0 < kPoolW; k0 += 4) {
    int ka = k0 + 2 * half;
    v2f a;
    a.x = P2[(mBase + l16) * kPoolW + ka];
    a.y = P2[(mBase + l16) * kPoolW + ka + 1];
    v2f b;
    b.x = Wout[(ka + 0) * kHID + nBase + l16];
    b.y = Wout[(ka + 1) * kHID + nBase + l16];
    acc = __builtin_amdgcn_wmma_f32_16x16x4_f32(
        /*neg_a=*/false, a, /*neg_b=*/false, b,
        /*c_mod=*/(short)0, acc, /*reuse_a=*/false, /*reuse_b=*/false);
  }

  float bias = bout[nBase + l16];
#pragma unroll
  for (int r = 0; r < 8; r++) {
    int m = mBase + r + 8 * half;  // C/D layout: VGPR r -> M=r (+8 for hi half)
    out[m * kHID + nBase + l16] = relu(acc[r] + bias);
  }
}

}  // namespace

// ---------------------------------------------------------------------------
extern "C" void kernel_launch(void* const* d_in, const int* in_sizes, int n_in,
                              void* d_out, int out_size, void* d_ws,
                              size_t ws_size, hipStream_t stream) {
  (void)in_sizes; (void)n_in; (void)out_size;
  const float* pts  = (const float*)d_in[0];
  const float* W1   = (const float*)d_in[1];
  const float* b1   = (const float*)d_in[2];
  const float* W2   = (const float*)d_in[3];
  const float* b2   = (const float*)d_in[4];
  const float* W3   = (const float*)d_in[5];
  const float* b3   = (const float*)d_in[6];
  const float* W4   = (const float*)d_in[7];
  const float* b4   = (const float*)d_in[8];
  const float* W5   = (const float*)d_in[9];
  const float* b5   = (const float*)d_in[10];
  const float* W6   = (const float*)d_in[11];
  const float* b6   = (const float*)d_in[12];
  const float* Wout = (const float*)d_in[13];
  const float* bout = (const float*)d_in[14];
  float* out = (float*)d_out;

  // Workspace layout: [gpool1 | gpool2 | h3 cache | type bytes]
  size_t poolBytes = (size_t)kTB * kPoolW * sizeof(unsigned);  // 320 KB each
  unsigned char* ws = (unsigned char*)d_ws;
  unsigned* gpool1 = (unsigned*)ws;
  unsigned* gpool2 = (unsigned*)(ws + poolBytes);
  float* h3buf = (float*)(ws + 2 * poolBytes);
  unsigned char* typebuf =
      (unsigned char*)(ws + 2 * poolBytes + (size_t)kTBN * kH * sizeof(float));
  size_t needFull =
      2 * poolBytes + (size_t)kTBN * kH * sizeof(float) + (size_t)kTBN;
  bool recompute = (ws_size < needFull);  // deterministic per ws_size

  // 0) zero both pool accumulators (contiguous)
  int poolWords = 2 * kTB * kPoolW;
  k_zero<<<(poolWords + kBlk - 1) / kBlk, kBlk, 0, stream>>>(gpool1, poolWords);

  // 1) per-point front MLP + first pooling
  dim3 grid1((unsigned)(kTBN / kBlk));
  if (recompute) {
    k_phase1<false><<<grid1, kBlk, 0, stream>>>(pts, W1, b1, W2, b2, W3, b3,
                                                h3buf, typebuf, gpool1);
  } else {
    k_phase1<true><<<grid1, kBlk, 0, stream>>>(pts, W1, b1, W2, b2, W3, b3,
                                               h3buf, typebuf, gpool1);
  }

  // 2) concat + back MLP + second pooling
  if (recompute) {
    k_phase2<true><<<grid1, kBlk, 0, stream>>>(
        pts, W1, b1, W2, b2, W3, b3, W4, b4, W5, b5, W6, b6, h3buf, typebuf,
        gpool1, gpool2);
  } else {
    k_phase2<false><<<grid1, kBlk, 0, stream>>>(
        pts, W1, b1, W2, b2, W3, b3, W4, b4, W5, b5, W6, b6, h3buf, typebuf,
        gpool1, gpool2);
  }

  // 3) WMMA output GEMM: 8 N-tiles x 32 M-tiles, one wave each
  dim3 grid3(kHID / 16, kTB / 16);
  k_out_gemm<<<grid3, 32, 0, stream>>>(gpool2, Wout, bout, out);
}